// Voxelization_87136296501765
// MI455X (gfx1250) — compile-verified
//
#include <hip/hip_runtime.h>
#include <stdint.h>

// ---------------- problem constants (from reference) ----------------
#define GXD 1504            // round((75.2-(-75.2))/0.1)
#define GYD 1504
#define GZD 40              // round((4-(-2))/0.15)
#define MAX_VOXELS 150000
#define MAX_POINTS 10
#define PC 5
#define VOX_ELEMS (MAX_VOXELS * MAX_POINTS * PC)   // 7,500,000

// ---------------- hash table / scan config ----------------
#define HBITS 23
#define HSIZE (1 << HBITS)        // 8,388,608 slots (>4x expected distinct voxels)
#define HMASK (HSIZE - 1)
#define TPB 256
#define SCAN_TILE 2048            // elements per scan block
#define SCAN_ITEMS 8              // per thread (256 threads * 8)

typedef unsigned int u32x4 __attribute__((ext_vector_type(4)));
typedef int          i32x4 __attribute__((ext_vector_type(4)));
typedef int          i32x8 __attribute__((ext_vector_type(8)));

// -------------------------------------------------------------------
// TDM: DMA `valid_elems` dwords from global into LDS (tile of
// `tile_elems` dwords; OOB reads return zero). One issue per wave.
// -------------------------------------------------------------------
__device__ __forceinline__ void tdm_load_dwords_to_lds(uint32_t lds_byte_off,
                                                       const void* gptr,
                                                       uint32_t valid_elems,
                                                       uint32_t tile_elems) {
#if __has_builtin(__builtin_amdgcn_tensor_load_to_lds)
    uint64_t ga = (uint64_t)(uintptr_t)gptr;
    u32x4 g0;
    g0[0] = 1u;                                            // count=1, user mode
    g0[1] = lds_byte_off;                                  // lds_addr (bytes)
    g0[2] = (uint32_t)ga;                                  // global_addr[31:0]
    g0[3] = (uint32_t)((ga >> 32) & 0x01FFFFFFu) | (2u << 30); // addr[56:32] | type=2
    i32x8 g1;
    g1[0] = (int)(2u << 16);                               // wg_mask=0, data_size=4B
    g1[1] = (int)((valid_elems & 0xFFFFu) << 16);          // tensor_dim0[15:0]
    g1[2] = (int)((valid_elems >> 16) & 0xFFFFu);          // tensor_dim0[31:16], dim1=0
    g1[3] = (int)((tile_elems & 0xFFFFu) << 16);           // tile_dim0
    g1[4] = 0;                                             // tile_dim1/2 = 0 (1D tile)
    g1[5] = (int)tile_elems;                               // tensor_dim0_stride lo
    g1[6] = 0;
    g1[7] = 0;
    i32x4 g2 = {0, 0, 0, 0};
    i32x4 g3 = {0, 0, 0, 0};
#if defined(__clang_major__) && (__clang_major__ >= 23)
    i32x8 g4 = {0, 0, 0, 0, 0, 0, 0, 0};
    __builtin_amdgcn_tensor_load_to_lds(g0, g1, g2, g3, g4, 0);
#else
    __builtin_amdgcn_tensor_load_to_lds(g0, g1, g2, g3, 0);
#endif
#if __has_builtin(__builtin_amdgcn_s_wait_tensorcnt)
    __builtin_amdgcn_s_wait_tensorcnt(0);
#else
    asm volatile("s_wait_tensorcnt 0x0" ::: "memory");
#endif
#else
    (void)lds_byte_off; (void)gptr; (void)valid_elems; (void)tile_elems;
#endif
}

// ---------------- init kernels ----------------
__global__ void voxel_init_tables_kernel(int* __restrict__ tab_key,
                                         int* __restrict__ tab_min,
                                         unsigned char* __restrict__ assigned, int n) {
    int i = blockIdx.x * blockDim.x + threadIdx.x;
    if (i < HSIZE) { tab_key[i] = -1; tab_min[i] = 0x7FFFFFFF; }
    if (i < n) assigned[i] = 0;
}

__global__ void voxel_init_out_kernel(float* __restrict__ out_vox,
                                      int* __restrict__ out_coors,
                                      int* __restrict__ out_num,
                                      int* __restrict__ out_vnum) {
    int i = blockIdx.x * blockDim.x + threadIdx.x;
    if (i < VOX_ELEMS) out_vox[i] = 0.0f;
    if (i < MAX_VOXELS * 3) out_coors[i] = -1;
    if (i < MAX_VOXELS) out_num[i] = 0;
    if (i == 0) out_vnum[0] = 0;
}

// ---------------- pass 1: bin points into hash table ----------------
__global__ void voxel_bin_kernel(const float* __restrict__ pts, int n,
                                 int* __restrict__ tab_key, int* __restrict__ tab_min,
                                 int* __restrict__ key_pt, int* __restrict__ slot_pt) {
    int i = blockIdx.x * blockDim.x + threadIdx.x;
    if (i >= n) return;
    const float* p = pts + (size_t)i * PC;
    float x = p[0], y = p[1], z = p[2];
    int cx = (int)floorf((x + 75.2f) / 0.1f);
    int cy = (int)floorf((y + 75.2f) / 0.1f);
    int cz = (int)floorf((z + 2.0f) / 0.15f);
    bool valid = (cx >= 0) & (cx < GXD) & (cy >= 0) & (cy < GYD) & (cz >= 0) & (cz < GZD);
    if (!valid) { slot_pt[i] = -1; key_pt[i] = -1; return; }
    int key = (cz * GYD + cy) * GXD + cx;          // < 90,480,640 (fits int32)
    uint32_t s = ((uint32_t)key * 2654435761u) & HMASK;
    while (true) {
        int k = atomicCAS(&tab_key[s], -1, key);
        if (k == -1 || k == key) break;
        s = (s + 1) & HMASK;
    }
    atomicMin(&tab_min[s], i);                     // deterministic first point per voxel
    key_pt[i] = key;
    slot_pt[i] = (int)s;
}

// ---------------- pass 2: first-occurrence flags ----------------
__global__ void voxel_flag_kernel(const int* __restrict__ slot_pt,
                                  const int* __restrict__ tab_min,
                                  int* __restrict__ flag_pt, int n) {
    int i = blockIdx.x * blockDim.x + threadIdx.x;
    if (i >= n) return;
    int s = slot_pt[i];
    flag_pt[i] = (s >= 0 && tab_min[s] == i) ? 1 : 0;
}

// ---------------- pass 3a: per-block exclusive scan (TDM -> LDS) ----------------
__global__ void __launch_bounds__(256)
voxel_scan_block_kernel(const int* __restrict__ flag, int* __restrict__ scan,
                        int* __restrict__ partials, int n) {
    __shared__ int tile[SCAN_TILE];
    __shared__ int sums[256];
    const int t = threadIdx.x;
    const int base = blockIdx.x * SCAN_TILE;

#if __has_builtin(__builtin_amdgcn_tensor_load_to_lds)
    // Each of the 8 wave32s DMAs its own 256-dword slice into LDS via the TDM.
    {
        int wave = t >> 5;
        int wbase = base + wave * 256;
        int rem = n - wbase;
        if (rem > 0) {
            uint32_t ve = (rem >= 256) ? 256u : (uint32_t)rem;
            uint32_t lds_off = (uint32_t)(uintptr_t)(&tile[0]) + (uint32_t)wave * 256u * 4u;
            tdm_load_dwords_to_lds(lds_off, flag + wbase, ve, 256u);
        } else {
#pragma unroll
            for (int k = 0; k < SCAN_ITEMS; ++k) tile[t * SCAN_ITEMS + k] = 0;
        }
    }
#else
#pragma unroll
    for (int k = 0; k < SCAN_ITEMS; ++k) {
        int idx = t * SCAN_ITEMS + k;
        tile[idx] = (base + idx < n) ? flag[base + idx] : 0;
    }
#endif
    __syncthreads();

    int v[SCAN_ITEMS];
    int local = 0;
#pragma unroll
    for (int k = 0; k < SCAN_ITEMS; ++k) { v[k] = tile[t * SCAN_ITEMS + k]; local += v[k]; }
    sums[t] = local;
    __syncthreads();
    for (int off = 1; off < 256; off <<= 1) {       // Hillis-Steele inclusive scan
        int x = (t >= off) ? sums[t - off] : 0;
        __syncthreads();
        sums[t] += x;
        __syncthreads();
    }
    int acc = sums[t] - local;                      // exclusive offset for this thread
#pragma unroll
    for (int k = 0; k < SCAN_ITEMS; ++k) {
        int idx = base + t * SCAN_ITEMS + k;
        if (idx < n) scan[idx] = acc;
        acc += v[k];
    }
    if (t == 255) partials[blockIdx.x] = sums[255];
}

// ---------------- pass 3b: scan block partials (single block) ----------------
__global__ void __launch_bounds__(1024)
voxel_scan_partials_kernel(int* __restrict__ partials, int nb, int* __restrict__ out_vnum) {
    __shared__ int s[1024];
    int t = threadIdx.x;
    int v = (t < nb) ? partials[t] : 0;
    s[t] = v;
    __syncthreads();
    for (int off = 1; off < 1024; off <<= 1) {
        int x = (t >= off) ? s[t - off] : 0;
        __syncthreads();
        s[t] += x;
        __syncthreads();
    }
    if (t < nb) partials[t] = s[t] - v;             // exclusive
    if (t == 1023) {
        int tot = s[1023];
        out_vnum[0] = (tot < MAX_VOXELS) ? tot : MAX_VOXELS;
    }
}

// ---------------- pass 4: assign voxel ids, write coors, reset curA ----------------
__global__ void voxel_assign_kernel(const int* __restrict__ slot_pt,
                                    const int* __restrict__ key_pt,
                                    const int* __restrict__ flag_pt,
                                    const int* __restrict__ scan_pt,
                                    const int* __restrict__ partials,
                                    int* __restrict__ tab_min,     // reused as cur-min buf 0
                                    int* __restrict__ tab_vid,
                                    int* __restrict__ out_coors, int n) {
    int i = blockIdx.x * blockDim.x + threadIdx.x;
    if (i >= n) return;
    int s = slot_pt[i];
    if (s < 0) return;
    tab_min[s] = 0x7FFFFFFF;                        // reset rank-pass-0 min buffer
    if (flag_pt[i]) {
        int v = scan_pt[i] + partials[i >> 11];     // 2048-wide scan blocks
        tab_vid[s] = (v < MAX_VOXELS) ? v : MAX_VOXELS;
        if (v < MAX_VOXELS) {
            int key = key_pt[i];
            int cx = key % GXD;
            int t2 = key / GXD;
            int cy = t2 % GYD;
            int cz = t2 / GYD;
            out_coors[3 * v + 0] = cz;              // stored (z, y, x)
            out_coors[3 * v + 1] = cy;
            out_coors[3 * v + 2] = cx;
        }
    }
}

// ---------------- rank round r: min then claim ----------------
__global__ void voxel_rank_min_kernel(const int* __restrict__ slot_pt,
                                      const unsigned char* __restrict__ assigned,
                                      int* __restrict__ cur, int n) {
    int i = blockIdx.x * blockDim.x + threadIdx.x;
    if (i >= n) return;
    int s = slot_pt[i];
    if (s < 0 || assigned[i]) return;
    atomicMin(&cur[s], i);                          // deterministic: smallest unassigned idx
}

__global__ void voxel_rank_claim_kernel(const float* __restrict__ pts,
                                        const int* __restrict__ slot_pt,
                                        unsigned char* __restrict__ assigned,
                                        const int* __restrict__ cur,
                                        int* __restrict__ nxt,
                                        const int* __restrict__ tab_vid,
                                        float* __restrict__ out_vox,
                                        int* __restrict__ out_num, int r, int n) {
    int i = blockIdx.x * blockDim.x + threadIdx.x;
    if (i >= n) return;
    int s = slot_pt[i];
    if (s < 0 || assigned[i]) return;
    const float* p = pts + (size_t)i * PC;
    __builtin_prefetch(p, 0, 0);                    // global_prefetch_b8 hint
    int m = cur[s];
    nxt[s] = 0x7FFFFFFF;                            // reset ping-pong buffer for round r+1
    if (m == i) {                                   // this point owns rank r of its voxel
        assigned[i] = 1;
        int v = tab_vid[s];
        if (v < MAX_VOXELS) {
            float* dst = out_vox + ((size_t)v * MAX_POINTS + r) * PC;
#pragma unroll
            for (int k = 0; k < PC; ++k) dst[k] = p[k];
            out_num[v] = r + 1;                     // monotone across rounds => min(cnt,10)
        }
    }
}

// -------------------------------------------------------------------
extern "C" void kernel_launch(void* const* d_in, const int* in_sizes, int n_in,
                              void* d_out, int out_size, void* d_ws, size_t ws_size,
                              hipStream_t stream) {
    (void)n_in; (void)out_size; (void)ws_size;
    const float* pts = (const float*)d_in[0];
    const int n = in_sizes[0] / PC;                 // 2,000,000

    // Output layout (floats, int regions bit-cast): voxels | coors | num | voxel_num
    float* out_vox  = (float*)d_out;
    int*   out_coors = (int*)(out_vox + VOX_ELEMS);
    int*   out_num   = out_coors + MAX_VOXELS * 3;
    int*   out_vnum  = out_num + MAX_VOXELS;

    // Workspace carve-up (~135 MB for n=2M)
    int* tab_key = (int*)d_ws;                      // HSIZE  (later: cur-min buf 1)
    int* tab_min = tab_key + HSIZE;                 // HSIZE  (later: cur-min buf 0)
    int* tab_vid = tab_min + HSIZE;                 // HSIZE
    int* key_pt  = tab_vid + HSIZE;                 // n
    int* slot_pt = key_pt + n;                      // n
    int* flag_pt = slot_pt + n;                     // n
    int* scan_pt = flag_pt + n;                     // n
    int* partials = scan_pt + n;                    // <=4096
    unsigned char* assigned = (unsigned char*)(partials + 4096); // n bytes

    const int nb_pts  = (n + TPB - 1) / TPB;
    const int nb_init = (HSIZE + TPB - 1) / TPB;
    const int nb_out  = (VOX_ELEMS + TPB - 1) / TPB;
    const int nb_scan = (n + SCAN_TILE - 1) / SCAN_TILE;   // 977 <= 1024

    voxel_init_tables_kernel<<<nb_init, TPB, 0, stream>>>(tab_key, tab_min, assigned, n);
    voxel_init_out_kernel<<<nb_out, TPB, 0, stream>>>(out_vox, out_coors, out_num, out_vnum);
    voxel_bin_kernel<<<nb_pts, TPB, 0, stream>>>(pts, n, tab_key, tab_min, key_pt, slot_pt);
    voxel_flag_kernel<<<nb_pts, TPB, 0, stream>>>(slot_pt, tab_min, flag_pt, n);
    voxel_scan_block_kernel<<<nb_scan, 256, 0, stream>>>(flag_pt, scan_pt, partials, n);
    voxel_scan_partials_kernel<<<1, 1024, 0, stream>>>(partials, nb_scan, out_vnum);
    voxel_assign_kernel<<<nb_pts, TPB, 0, stream>>>(slot_pt, key_pt, flag_pt, scan_pt,
                                                    partials, tab_min, tab_vid, out_coors, n);
    int* bufs[2] = {tab_min, tab_key};
    for (int r = 0; r < MAX_POINTS; ++r) {
        int* cur = bufs[r & 1];
        int* nxt = bufs[(r + 1) & 1];
        voxel_rank_min_kernel<<<nb_pts, TPB, 0, stream>>>(slot_pt, assigned, cur, n);
        voxel_rank_claim_kernel<<<nb_pts, TPB, 0, stream>>>(pts, slot_pt, assigned, cur, nxt,
                                                            tab_vid, out_vox, out_num, r, n);
    }
}